// SimpleGCNConv_26431228739923
// MI455X (gfx1250) — compile-verified
//
#include <hip/hip_runtime.h>
#include <hip/hip_bf16.h>

typedef float v2f __attribute__((ext_vector_type(2)));
typedef float v8f __attribute__((ext_vector_type(8)));

#define N_NODES 8192
#define C_DIM   256
#define N_EDGES 131072
#define ROW_WORDS (N_NODES / 32)   // 256 u32 words per bitmap row

// ---------------- K0: zero the adjacency bitmap ----------------
__global__ void zero_bitmap_kernel(uint4* __restrict__ p, int n4) {
    int i = blockIdx.x * blockDim.x + threadIdx.x;
    int stride = gridDim.x * blockDim.x;
    uint4 z = {0u, 0u, 0u, 0u};
    for (; i < n4; i += stride) p[i] = z;
}

// ---------------- K1: scatter edges into bitmap (dedup via atomicOr) ----------------
__global__ void scatter_edges_kernel(const long long* __restrict__ ei,
                                     unsigned* __restrict__ bitmap) {
    int i = blockIdx.x * blockDim.x + threadIdx.x;
    if (i >= N_EDGES) return;
    int r = (int)ei[i];            // edge_index[0][i]
    int c = (int)ei[N_EDGES + i];  // edge_index[1][i]
    atomicOr(&bitmap[r * ROW_WORDS + (c >> 5)], 1u << (c & 31));
}

// ---------------- K2: degree + D^{-1/2}  (one wave per row) ----------------
__global__ void degree_kernel(const unsigned* __restrict__ bitmap,
                              float* __restrict__ dis) {
    int row  = blockIdx.x * (blockDim.x >> 5) + (threadIdx.x >> 5);
    int lane = threadIdx.x & 31;
    const unsigned* rb = bitmap + row * ROW_WORDS;
    int pc = 0;
#pragma unroll
    for (int t = 0; t < 8; ++t) pc += __popc(rb[lane + 32 * t]);
#pragma unroll
    for (int off = 16; off > 0; off >>= 1) pc += __shfl_xor(pc, off, 32);
    if (lane == 0) dis[row] = rsqrtf((float)(pc + 1));  // +1 from diagonal add; deg >= 1
}

// ---------------- K3: y = x @ W^T via fp32 WMMA 16x16x4 ----------------
// One wave computes one 16x16 output tile, iterating K=256 in steps of 4.
// A (16x4 f32) layout: lanes 0-15 hold M=lane, K in {0,1}; lanes 16-31 hold M=lane-16, K in {2,3}.
// B (4x16 f32) layout: VGPR v holds K=v (lanes 0-15) / K=v+2 (lanes 16-31), N = lane&15.
// Both are a contiguous float2 load per lane from row-major x / W.
__global__ void gemm_xwT_kernel(const float* __restrict__ x,
                                const float* __restrict__ W,
                                float* __restrict__ y) {
    int wave = (blockIdx.x * blockDim.x + threadIdx.x) >> 5;  // 0..8191
    int lane = threadIdx.x & 31;
    int tileM = wave >> 4;   // 0..511
    int tileN = wave & 15;   // 0..15
    int m0 = tileM * 16, n0 = tileN * 16;
    int half = lane >> 4;    // 0 or 1
    int l15  = lane & 15;

    const float* arow = x + (size_t)(m0 + l15) * C_DIM + 2 * half;  // x[m0+l15][k + 2*half ..]
    const float* brow = W + (size_t)(n0 + l15) * C_DIM + 2 * half;  // W[n0+l15][k + 2*half ..] == B[k..][n0+l15]

    v8f acc = {};
#pragma unroll 4
    for (int k = 0; k < C_DIM; k += 4) {
        v2f a = *(const v2f*)(arow + k);
        v2f b = *(const v2f*)(brow + k);
        acc = __builtin_amdgcn_wmma_f32_16x16x4_f32(
            /*neg_a=*/false, a, /*neg_b=*/false, b,
            /*c_mod=*/(short)0, acc, /*reuse_a=*/false, /*reuse_b=*/false);
    }
    // D layout: VGPR v -> row m0 + v + 8*half, col n0 + l15
#pragma unroll
    for (int v = 0; v < 8; ++v) {
        y[(size_t)(m0 + v + 8 * half) * C_DIM + (n0 + l15)] = acc[v];
    }
}

// ---------------- K4: sparse aggregation  out = D^-1/2 (A + I) D^-1/2 y + b ----------------
// One wave per row; 32 lanes x 8 regs = 256 channels held in registers.
__global__ void aggregate_kernel(const float* __restrict__ y,
                                 const unsigned* __restrict__ bitmap,
                                 const float* __restrict__ dis,
                                 const float* __restrict__ b_lin,
                                 const float* __restrict__ bias,
                                 float* __restrict__ out) {
    int row  = blockIdx.x * (blockDim.x >> 5) + (threadIdx.x >> 5);
    int lane = threadIdx.x & 31;
    float di = dis[row];

    float acc[8];
    const float* yr = y + (size_t)row * C_DIM;
#pragma unroll
    for (int t = 0; t < 8; ++t) acc[t] = di * yr[lane + 32 * t];  // diagonal (+I) term

    const unsigned* rb = bitmap + row * ROW_WORDS;
    for (int w = 0; w < ROW_WORDS; ++w) {
        unsigned bits = (unsigned)__builtin_amdgcn_readfirstlane((int)rb[w]);
        while (bits) {
            int b = __builtin_ctz(bits);
            bits &= bits - 1;
            int j = (w << 5) + b;
            float s = dis[j];
            const float* yj = y + (size_t)j * C_DIM;
#pragma unroll
            for (int t = 0; t < 8; ++t) acc[t] = fmaf(s, yj[lane + 32 * t], acc[t]);
        }
    }
#pragma unroll
    for (int t = 0; t < 8; ++t) {
        int c = lane + 32 * t;
        out[(size_t)row * C_DIM + c] = fmaf(di, acc[t], b_lin[c] + bias[c]);
    }
}

extern "C" void kernel_launch(void* const* d_in, const int* in_sizes, int n_in,
                              void* d_out, int out_size, void* d_ws, size_t ws_size,
                              hipStream_t stream) {
    const float*     x     = (const float*)d_in[0];      // (N, C_IN)
    const float*     W     = (const float*)d_in[1];      // (C_OUT, C_IN)
    const float*     b_lin = (const float*)d_in[2];      // (C_OUT,)
    const float*     bias  = (const float*)d_in[3];      // (C_OUT,)
    const long long* ei    = (const long long*)d_in[4];  // (2, E) int64
    float* out = (float*)d_out;

    // Workspace layout
    char* ws = (char*)d_ws;
    unsigned* bitmap = (unsigned*)ws;                              // 8 MB
    float*    y      = (float*)(ws + (size_t)N_NODES * ROW_WORDS * 4);  // 8 MB
    float*    dis    = (float*)(ws + 2 * (size_t)N_NODES * ROW_WORDS * 4); // 32 KB

    // K0: zero bitmap (2M words = 512K uint4)
    {
        int n4 = (N_NODES * ROW_WORDS) / 4;
        zero_bitmap_kernel<<<2048, 256, 0, stream>>>((uint4*)bitmap, n4);
    }
    // K1: scatter edges
    scatter_edges_kernel<<<(N_EDGES + 255) / 256, 256, 0, stream>>>(ei, bitmap);
    // K2: degrees -> D^{-1/2}   (8 waves / block, 1 row / wave)
    degree_kernel<<<N_NODES / 8, 256, 0, stream>>>(bitmap, dis);
    // K3: y = x @ W^T   (8192 tiles, 8 waves / block)
    gemm_xwT_kernel<<<(N_NODES / 16) * (C_DIM / 16) / 8, 256, 0, stream>>>(x, W, y);
    // K4: sparse normalized aggregation + bias
    aggregate_kernel<<<N_NODES / 8, 256, 0, stream>>>(y, bitmap, dis, b_lin, bias, out);
}